// CCA_64201171141469
// MI455X (gfx1250) — compile-verified
//
#include <hip/hip_runtime.h>
#include <math.h>

// ---------------------------------------------------------------------------
// CCA correlation, MI455X (gfx1250), wave32 + WMMA f32 path.
//   corr^2 = tr(A^{-1} C B^{-1} C^T),  A=Sigma11, B=Sigma22, C=Sigma12
//   A = La La^T, B = Lb Lb^T ; U = La^{-1} C ; V = Lb^{-1} U^T ; out=-||V||_F
// Gram kernel: raw Gram via WMMA f32, mean correction folded into epilogue:
//   Sigma = sp*(Sum xy - M*mx*my) (+ r on diag)
// Staging: GLOBAL_LOAD_ASYNC_TO_LDS_B128 double-buffered (ASYNCcnt pipelining)
// ---------------------------------------------------------------------------

typedef float v2f __attribute__((ext_vector_type(2)));
typedef float v8f __attribute__((ext_vector_type(8)));

#define MROWS 16384
#define NF    2048
#define OO    1024
#define KB    32
#define LSTR  144   // LDS row stride (floats): (2 rows)*144 = 288 -> bank +32
#define NCHUNK (MROWS / KB)

#if defined(__gfx1250__) && \
    __has_builtin(__builtin_amdgcn_global_load_async_to_lds_b128) && \
    __has_builtin(__builtin_amdgcn_s_wait_asynccnt)
#define HAVE_ASYNC_LDS 1
#define GLOBAL_AS __attribute__((address_space(1)))
#define LDS_AS    __attribute__((address_space(3)))
typedef int v4i __attribute__((__vector_size__(4 * sizeof(int))));
#else
#define HAVE_ASYNC_LDS 0
#endif

// ---------------- column means ----------------
__global__ void mean_kernel(const float* __restrict__ X, float* __restrict__ mean) {
  int f = blockIdx.x * blockDim.x + threadIdx.x;   // 0..2047
  const float* p = X + f;
  float s = 0.f;
#pragma unroll 8
  for (int k = 0; k < MROWS; ++k) s += p[(size_t)k * NF];
  mean[f] = s * (1.0f / (float)MROWS);
}

// ---------------- shared WMMA 128x128 microkernel (per chunk) --------------
__device__ __forceinline__ void wmma_chunk(const float* __restrict__ As,
                                           const float* __restrict__ Bs,
                                           int aoff, int boff, int lm, int lk,
                                           v8f acc[4][4]) {
#pragma unroll
  for (int ks = 0; ks < KB; ks += 4) {
    v2f af[4], bf[4];
#pragma unroll
    for (int i = 0; i < 4; ++i) {
      af[i].x = As[(ks + lk) * LSTR + aoff + i * 16 + lm];
      af[i].y = As[(ks + lk + 1) * LSTR + aoff + i * 16 + lm];
      bf[i].x = Bs[(ks + lk) * LSTR + boff + i * 16 + lm];
      bf[i].y = Bs[(ks + lk + 1) * LSTR + boff + i * 16 + lm];
    }
#pragma unroll
    for (int i = 0; i < 4; ++i)
#pragma unroll
      for (int j = 0; j < 4; ++j)
        acc[i][j] = __builtin_amdgcn_wmma_f32_16x16x4_f32(
            false, af[i], false, bf[j], (short)0, acc[i][j], false, false);
  }
}

// ---------------- Gram blocks: SA=S11, SB=S22, SC=S12 ----------------------
// Async path: double-buffered LDS, GLOBAL_LOAD_ASYNC_TO_LDS_B128 copies of
// chunk k+1 overlap the 128 WMMAs of chunk k (ASYNCcnt pipelining).
__global__ void __launch_bounds__(128)
gram_wmma_kernel(const float* __restrict__ X, const float* __restrict__ mean,
                 float* __restrict__ SA, float* __restrict__ SB,
                 float* __restrict__ SC) {
  extern __shared__ float smem[];   // 4 * KB * LSTR floats (2 x (As,Bs))

  const int z = blockIdx.z;
  const int rowOff = (z == 1) ? OO : 0;
  const int colOff = (z == 0) ? 0 : OO;
  float* out = (z == 0) ? SA : ((z == 1) ? SB : SC);
  const bool addDiag = (z < 2);

  const int m0 = blockIdx.y * 128;
  const int n0 = blockIdx.x * 128;
  const int tid  = threadIdx.x;
  const int lane = tid & 31;
  const int wave = tid >> 5;
  const int aoff = (wave >> 1) * 64;
  const int boff = (wave & 1) * 64;
  const int lm = lane & 15;
  const int lk = (lane >> 4) << 1;

  float* As0 = smem;
  float* Bs0 = smem + KB * LSTR;
  float* As1 = smem + 2 * KB * LSTR;
  float* Bs1 = smem + 3 * KB * LSTR;

  v8f acc[4][4];
  const v8f vz = {};
#pragma unroll
  for (int i = 0; i < 4; ++i)
#pragma unroll
    for (int j = 0; j < 4; ++j) acc[i][j] = vz;

#if HAVE_ASYNC_LDS
  // thread -> (4-float column group, row group); 16 b128 copies per chunk
  const int c4  = (tid & 31) * 4;
  const int rB  = tid >> 5;                 // 0..3
  const float* gAp = X + rowOff + m0 + c4;
  const float* gBp = X + colOff + n0 + c4;

  auto issue = [&](int kb, int buf) {
    float* As = buf ? As1 : As0;
    float* Bs = buf ? Bs1 : Bs0;
#pragma unroll
    for (int rr = 0; rr < KB; rr += 4) {
      const int r = rB + rr;
      __builtin_amdgcn_global_load_async_to_lds_b128(
          (GLOBAL_AS v4i*)(gAp + (size_t)(kb + r) * NF),
          (LDS_AS v4i*)(As + r * LSTR + c4), 0, 0);
      __builtin_amdgcn_global_load_async_to_lds_b128(
          (GLOBAL_AS v4i*)(gBp + (size_t)(kb + r) * NF),
          (LDS_AS v4i*)(Bs + r * LSTR + c4), 0, 0);
    }
  };

  issue(0, 0);
  int k = 0;
  for (; k < NCHUNK - 1; ++k) {
    const int cur = k & 1;
    issue((k + 1) * KB, cur ^ 1);                    // prefetch next chunk
    __builtin_amdgcn_s_wait_asynccnt(16);            // chunk k copies landed
    __syncthreads();                                 // ... for all waves
    wmma_chunk(cur ? As1 : As0, cur ? Bs1 : Bs0, aoff, boff, lm, lk, acc);
    __syncthreads();                                 // buffer free for reuse
  }
  __builtin_amdgcn_s_wait_asynccnt(0);
  __syncthreads();
  wmma_chunk((k & 1) ? As1 : As0, (k & 1) ? Bs1 : Bs0, aoff, boff, lm, lk, acc);
#else
  // synchronous fallback staging (single buffer)
  const int c = tid;
  const float* gA = X + rowOff + m0 + c;
  const float* gB = X + colOff + n0 + c;
  for (int kb = 0; kb < MROWS; kb += KB) {
    __syncthreads();
#pragma unroll
    for (int r = 0; r < KB; ++r) {
      As0[r * LSTR + c] = gA[(size_t)(kb + r) * NF];
      Bs0[r * LSTR + c] = gB[(size_t)(kb + r) * NF];
    }
    __syncthreads();
    wmma_chunk(As0, Bs0, aoff, boff, lm, lk, acc);
  }
#endif

  // epilogue: Sigma = sp * (acc - M*mx*my) (+ r on diagonal of S11/S22)
  const float sp = 1.0f / (float)(MROWS - 1);
  const float Mf = (float)MROWS;
  const int halfSel = (lane >> 4) << 3;  // 0 or 8
#pragma unroll
  for (int j = 0; j < 4; ++j) {
    const int col = n0 + boff + j * 16 + lm;
    const float mcol = mean[colOff + col];
#pragma unroll
    for (int i = 0; i < 4; ++i) {
      const int rb = m0 + aoff + i * 16 + halfSel;
#pragma unroll
      for (int v = 0; v < 8; ++v) {
        const int row = rb + v;
        const float mrow = mean[rowOff + row];
        float val = sp * (acc[i][j][v] - Mf * mrow * mcol);
        if (addDiag && (row == col)) val += 1e-4f;
        out[(size_t)row * OO + col] = val;
      }
    }
  }
}

// ---------------- generic C += alpha * op(A) op(B), 128x128 per block ------
// transX==0: operand stored k-major   (elem[k][m] = P[k*ld + m0+m])
// transX==1: operand stored k-minor   (elem[k][m] = P[(m0+m)*ld + k])
__global__ void __launch_bounds__(128)
wmma_gemm_acc_kernel(float* __restrict__ C, int ldc,
                     const float* __restrict__ A, int lda, int transA,
                     const float* __restrict__ B, int ldb, int transB,
                     int K, float alpha) {
  __shared__ float As[KB * LSTR];
  __shared__ float Bs[KB * LSTR];

  const int m0 = blockIdx.y * 128;
  const int n0 = blockIdx.x * 128;
  const int tid  = threadIdx.x;
  const int lane = tid & 31;
  const int wave = tid >> 5;
  const int aoff = (wave >> 1) * 64;
  const int boff = (wave & 1) * 64;
  const int lm = lane & 15;
  const int lk = (lane >> 4) << 1;

  v8f acc[4][4];
  const v8f vz = {};
#pragma unroll
  for (int i = 0; i < 4; ++i)
#pragma unroll
    for (int j = 0; j < 4; ++j) acc[i][j] = vz;

  for (int kb = 0; kb < K; kb += KB) {
    __syncthreads();
    if (transA == 0) {
      for (int it = tid; it < KB * 128; it += 128) {
        const int r = it >> 7, cc = it & 127;
        As[r * LSTR + cc] = A[(size_t)(kb + r) * lda + m0 + cc];
      }
    } else {
      for (int it = tid; it < KB * 128; it += 128) {
        const int cc = it >> 5, r = it & 31;
        As[r * LSTR + cc] = A[(size_t)(m0 + cc) * lda + kb + r];
      }
    }
    if (transB == 0) {
      for (int it = tid; it < KB * 128; it += 128) {
        const int r = it >> 7, cc = it & 127;
        Bs[r * LSTR + cc] = B[(size_t)(kb + r) * ldb + n0 + cc];
      }
    } else {
      for (int it = tid; it < KB * 128; it += 128) {
        const int cc = it >> 5, r = it & 31;
        Bs[r * LSTR + cc] = B[(size_t)(n0 + cc) * ldb + kb + r];
      }
    }
    __syncthreads();
    wmma_chunk(As, Bs, aoff, boff, lm, lk, acc);
  }

  const int halfSel = (lane >> 4) << 3;
#pragma unroll
  for (int i = 0; i < 4; ++i) {
#pragma unroll
    for (int j = 0; j < 4; ++j) {
      const int col = n0 + boff + j * 16 + lm;
      const int rb  = m0 + aoff + i * 16 + halfSel;
#pragma unroll
      for (int v = 0; v < 8; ++v) {
        const int row = rb + v;
        C[(size_t)row * ldc + col] += alpha * acc[i][j][v];
      }
    }
  }
}

// ---------------- Cholesky: factor 128x128 diagonal block (1 WG) -----------
__global__ void chol_diag_kernel(float* __restrict__ S, int k0) {
  extern __shared__ float Lb[];   // 128 * 129 floats (dynamic LDS)
  const int P = 129;
  const int tid = threadIdx.x;    // 128 threads, thread == row
  for (int it = tid; it < 128 * 128; it += 128) {
    const int r = it >> 7, c = it & 127;
    Lb[r * P + c] = S[(size_t)(k0 + r) * OO + k0 + c];
  }
  __syncthreads();
  for (int kk = 0; kk < 128; ++kk) {
    const float dv  = Lb[kk * P + kk];
    const float d   = sqrtf(dv);
    const float inv = 1.0f / d;
    __syncthreads();
    if (tid == kk)      Lb[kk * P + kk] = d;
    else if (tid > kk)  Lb[tid * P + kk] *= inv;
    __syncthreads();
    if (tid > kk) {
      const float lik = Lb[tid * P + kk];
      for (int c = kk + 1; c <= tid; ++c)
        Lb[tid * P + c] -= lik * Lb[c * P + kk];
    }
    __syncthreads();
  }
  for (int it = tid; it < 128 * 128; it += 128) {
    const int r = it >> 7, c = it & 127;
    S[(size_t)(k0 + r) * OO + k0 + c] = Lb[r * P + c];
  }
}

// ---------------- Cholesky panel: rows r0.. solve X * Lkk^T = A ------------
__global__ void __launch_bounds__(64)
panel_trsm_kernel(float* __restrict__ S, int r0, int k0) {
  __shared__ float xs[64][129];
  const int t   = threadIdx.x;
  const int row = r0 + blockIdx.x * 64 + t;
  float* rp = S + (size_t)row * OO + k0;
  const float* L = S + (size_t)k0 * OO + k0;   // diag block, ld = OO
  for (int c = 0; c < 128; ++c) xs[t][c] = rp[c];
  for (int j = 0; j < 128; ++j) {
    float s = xs[t][j];
    for (int c = 0; c < j; ++c) s -= L[(size_t)j * OO + c] * xs[t][c];
    xs[t][j] = s / L[(size_t)j * OO + j];
  }
  for (int c = 0; c < 128; ++c) rp[c] = xs[t][c];
}

// ---------------- RHS diag solve: Lkk * X = X (rows k0..k0+127, all cols) --
__global__ void __launch_bounds__(64)
trsm_diag_cols_kernel(const float* __restrict__ L, float* __restrict__ U, int k0) {
  __shared__ float xs[64][129];
  const int t   = threadIdx.x;
  const int col = blockIdx.x * 64 + t;
  const float* Lk = L + (size_t)k0 * OO + k0;
  for (int j = 0; j < 128; ++j) xs[t][j] = U[(size_t)(k0 + j) * OO + col];
  for (int j = 0; j < 128; ++j) {
    float s = xs[t][j];
    for (int c = 0; c < j; ++c) s -= Lk[(size_t)j * OO + c] * xs[t][c];
    xs[t][j] = s / Lk[(size_t)j * OO + j];
  }
  for (int j = 0; j < 128; ++j) U[(size_t)(k0 + j) * OO + col] = xs[t][j];
}

// ---------------- misc -----------------------------------------------------
__global__ void transpose_kernel(const float* __restrict__ src,
                                 float* __restrict__ dst) {
  const int id = blockIdx.x * blockDim.x + threadIdx.x;  // 0..1M-1
  const int i = id >> 10, j = id & 1023;
  dst[(size_t)id] = src[(size_t)j * OO + i];
}

__global__ void zero_kernel(float* p) { p[0] = 0.f; }

__global__ void reduce_sq_kernel(const float* __restrict__ V,
                                 float* __restrict__ acc) {
  __shared__ float sm[256];
  const int tid = threadIdx.x;
  const size_t base = (size_t)blockIdx.x * 1024;
  float s = 0.f;
  for (int i = tid; i < 1024; i += 256) { const float v = V[base + i]; s += v * v; }
  sm[tid] = s;
  __syncthreads();
  for (int off = 128; off > 0; off >>= 1) {
    if (tid < off) sm[tid] += sm[tid + off];
    __syncthreads();
  }
  if (tid == 0) atomicAdd(acc, sm[0]);
}

__global__ void finalize_kernel(const float* __restrict__ acc,
                                float* __restrict__ out) {
  out[0] = -sqrtf(acc[0]);
}

// ---------------------------------------------------------------------------
extern "C" void kernel_launch(void* const* d_in, const int* in_sizes, int n_in,
                              void* d_out, int out_size, void* d_ws, size_t ws_size,
                              hipStream_t stream) {
  const float* X = (const float*)d_in[0];
  float* out = (float*)d_out;
  float* ws  = (float*)d_ws;

  float* mean = ws;                       // 2048
  float* acc  = ws + 2048;                // 1
  float* SA   = ws + 4096;                // 1024*1024
  float* SB   = SA + (size_t)OO * OO;
  float* SC   = SB + (size_t)OO * OO;     // becomes U = La^{-1} C
  float* V    = SC + (size_t)OO * OO;     // becomes V = Lb^{-1} U^T

  mean_kernel<<<dim3(8), dim3(256), 0, stream>>>(X, mean);
  gram_wmma_kernel<<<dim3(8, 8, 3), dim3(128),
                     4 * KB * LSTR * sizeof(float), stream>>>(X, mean, SA, SB, SC);

  // Blocked Cholesky of SA and SB (lower), NB = 128
  for (int which = 0; which < 2; ++which) {
    float* S = which ? SB : SA;
    for (int k = 0; k < 8; ++k) {
      const int k0 = k * 128;
      chol_diag_kernel<<<dim3(1), dim3(128), 128 * 129 * sizeof(float), stream>>>(S, k0);
      const int nt = OO - (k + 1) * 128;
      if (nt > 0) {
        panel_trsm_kernel<<<dim3(nt / 64), dim3(64), 0, stream>>>(S, k0 + 128, k0);
        // trailing SYRK: S[t0:,t0:] -= P P^T,  P = S[t0:, k0:k0+128]
        wmma_gemm_acc_kernel<<<dim3(nt / 128, nt / 128), dim3(128), 0, stream>>>(
            S + (size_t)(k0 + 128) * OO + (k0 + 128), OO,
            S + (size_t)(k0 + 128) * OO + k0, OO, 1,
            S + (size_t)(k0 + 128) * OO + k0, OO, 1,
            128, -1.0f);
      }
    }
  }

  // U = La^{-1} C   (in place in SC), blocked forward substitution
  for (int k = 0; k < 8; ++k) {
    const int k0 = k * 128;
    if (k0 > 0) {
      wmma_gemm_acc_kernel<<<dim3(8, 1), dim3(128), 0, stream>>>(
          SC + (size_t)k0 * OO, OO,
          SA + (size_t)k0 * OO, OO, 1,   // L[k, 0:k0]  (k-minor)
          SC, OO, 0,                     // solved U rows (k-major)
          k0, -1.0f);
    }
    trsm_diag_cols_kernel<<<dim3(16), dim3(64), 0, stream>>>(SA, SC, k0);
  }

  // V0 = U^T ; V = Lb^{-1} V0 (in place in V)
  transpose_kernel<<<dim3(4096), dim3(256), 0, stream>>>(SC, V);
  for (int k = 0; k < 8; ++k) {
    const int k0 = k * 128;
    if (k0 > 0) {
      wmma_gemm_acc_kernel<<<dim3(8, 1), dim3(128), 0, stream>>>(
          V + (size_t)k0 * OO, OO,
          SB + (size_t)k0 * OO, OO, 1,
          V, OO, 0,
          k0, -1.0f);
    }
    trsm_diag_cols_kernel<<<dim3(16), dim3(64), 0, stream>>>(SB, V, k0);
  }

  // out = -sqrt(sum V^2)
  zero_kernel<<<1, 1, 0, stream>>>(acc);
  reduce_sq_kernel<<<dim3(1024), dim3(256), 0, stream>>>(V, acc);
  finalize_kernel<<<1, 1, 0, stream>>>(acc, out);
}